// MWNO_CZ_18751827214638
// MI455X (gfx1250) — compile-verified
//
#include <hip/hip_runtime.h>
#include <math.h>

typedef __attribute__((ext_vector_type(2))) float v2f;
typedef __attribute__((ext_vector_type(8))) float v8f;

#if defined(__has_builtin)
#if __has_builtin(__builtin_amdgcn_wmma_f32_16x16x4_f32)
#define MWNO_HAVE_WMMA_F32 1
#endif
#if __has_builtin(__builtin_amdgcn_s_wait_dscnt)
#define MWNO_DS_FENCE() __builtin_amdgcn_s_wait_dscnt(0)
#endif
#endif
#ifndef MWNO_DS_FENCE
#define MWNO_DS_FENCE() asm volatile("s_wait_dscnt 0" ::: "memory")
#endif

#define BQ 16
#define NQ 8192
#define CQ 32
#define KQ 4
#define CKQ 128
#define ALPHAQ 128
#define NLEV 13

// ---------------------------------------------------------------------------
// Batched strided fp32 GEMM on the WMMA pipe (v_wmma_f32_16x16x4_f32).
//   C[z; r, c] (+)= alpha * sum_k A[z; r, k] * B[z; k, c]
// A and B have contiguous columns (csA = csB = 1); C strides are generic.
// Block = 8 waves; each wave owns a 16x32 C tile (two 16x16 WMMA tiles that
// share one A fragment -> 8 WMMAs per 24 global loads per 16-K chunk).
// K is processed in chunks of 16:
//   - each wave stages its 16x16 A chunk into a private 1KB LDS slice with
//     row-contiguous (coalesced) global loads, then reads WMMA fragments back
//     as adjacent ds_load pairs (fixes the stride-rsA gather of direct loads)
//   - B is read directly (lane index = column -> already coalesced)
//   - M/N ragged edges use address *clamping* (those C rows/cols are never
//     stored); the K tail zero-fills A at stage time and masks B -> the hot
//     loop has no per-element predication and no EXEC churn around WMMAs.
// Wave-local LDS visibility only needs s_wait_dscnt 0 (same-wave DS ops are
// in-order); no block barrier, so early-exited waves are harmless.
// ---------------------------------------------------------------------------
__global__ __launch_bounds__(256) void mwno_gemm(
    const float* __restrict__ A, long rsA, long bsA,
    const float* __restrict__ B, long rsB, long bsB,
    float* __restrict__ C, long rsC, long csC, long bsC,
    int M, int N, int Kd, float alpha, int accum)
{
  const int z = blockIdx.z;
  A += (long)z * bsA;
  B += (long)z * bsB;
  C += (long)z * bsC;

  const int tilesN = (N + 31) >> 5;            // 32-wide wave tiles in N
  const int tilesM = (M + 15) >> 4;
  const int wave = threadIdx.x >> 5;
  const int lane = threadIdx.x & 31;
  const long tile = (long)blockIdx.x * 8 + wave;
  if (tile >= (long)tilesM * tilesN) return;   // wave-uniform exit
  const int tm = (int)(tile / tilesN);
  const int tn = (int)(tile % tilesN);
  const int row0 = tm << 4;
  const int col0 = tn << 5;

  const int mloc = lane & 15;                  // fragment row (A/D) / col (B/D)
  const int khalf = (lane >> 4) << 1;          // K sub-offset: 0 or 2

#if MWNO_HAVE_WMMA_F32
  __shared__ float AsLds[8 * 256];             // 1KB per wave

  // Staging geometry: lane pair per row, 8 contiguous floats per lane.
  const int srow = lane >> 1;
  const int shalf = (lane & 1) << 3;
  float* As = AsLds + wave * 256;

  // Clamped base pointers (out-of-range rows/cols produce unstored C entries).
  const int arow_i = (row0 + srow < M) ? (row0 + srow) : (M - 1);
  const int bc0 = (col0 + mloc < N)      ? (col0 + mloc)      : (N - 1);
  const int bc1 = (col0 + 16 + mloc < N) ? (col0 + 16 + mloc) : (N - 1);
  const float* arow = A + (long)arow_i * rsA;
  const float* afrag = As + mloc * 16 + khalf;
  const float* pb0 = B + bc0 + (long)khalf * rsB;
  const float* pb1 = B + bc1 + (long)khalf * rsB;
  const long rsB4 = 4 * rsB;

  v8f acc0 = {0.f, 0.f, 0.f, 0.f, 0.f, 0.f, 0.f, 0.f};
  v8f acc1 = {0.f, 0.f, 0.f, 0.f, 0.f, 0.f, 0.f, 0.f};

  int k0 = 0;
  for (; k0 + 16 <= Kd; k0 += 16) {
    // --- stage A chunk (coalesced, unconditional) ---
    const float* p = arow + k0 + shalf;
#pragma unroll
    for (int i = 0; i < 8; ++i) As[srow * 16 + shalf + i] = p[i];
    MWNO_DS_FENCE();
    // --- 8 x wmma on the chunk (A fragment reused across both N tiles) ---
#pragma unroll
    for (int kk = 0; kk < 16; kk += 4) {
      v2f a, b0, b1;
      a.x = afrag[kk];
      a.y = afrag[kk + 1];
      b0.x = pb0[0];
      b0.y = pb0[rsB];
      b1.x = pb1[0];
      b1.y = pb1[rsB];
      pb0 += rsB4;
      pb1 += rsB4;
      acc0 = __builtin_amdgcn_wmma_f32_16x16x4_f32(false, a, false, b0,
                                                   (short)0, acc0, false, false);
      acc1 = __builtin_amdgcn_wmma_f32_16x16x4_f32(false, a, false, b1,
                                                   (short)0, acc1, false, false);
    }
  }

  if (k0 < Kd) {
    // --- tail chunk: zero-fill A beyond Kd, clamp+mask B ---
#pragma unroll
    for (int i = 0; i < 8; ++i) {
      const int k = k0 + shalf + i;
      const int kc = (k < Kd) ? k : (Kd - 1);
      const float v = arow[kc];
      As[srow * 16 + shalf + i] = (k < Kd) ? v : 0.f;
    }
    MWNO_DS_FENCE();
#pragma unroll
    for (int kk = 0; kk < 16; kk += 4) {
      const int ka = k0 + kk + khalf;
      v2f a, b0, b1;
      a.x = afrag[kk];
      a.y = afrag[kk + 1];
      const long ka0 = (ka < Kd) ? ka : (Kd - 1);
      const long ka1 = (ka + 1 < Kd) ? (ka + 1) : (Kd - 1);
      const float m0 = (ka < Kd) ? 1.f : 0.f;
      const float m1 = (ka + 1 < Kd) ? 1.f : 0.f;
      b0.x = B[ka0 * rsB + bc0] * m0;
      b0.y = B[ka1 * rsB + bc0] * m1;
      b1.x = B[ka0 * rsB + bc1] * m0;
      b1.y = B[ka1 * rsB + bc1] * m1;
      acc0 = __builtin_amdgcn_wmma_f32_16x16x4_f32(false, a, false, b0,
                                                   (short)0, acc0, false, false);
      acc1 = __builtin_amdgcn_wmma_f32_16x16x4_f32(false, a, false, b1,
                                                   (short)0, acc1, false, false);
    }
  }
#else
  // Scalar fallback with the same per-lane D ownership (insurance only).
  v8f acc0 = {0.f, 0.f, 0.f, 0.f, 0.f, 0.f, 0.f, 0.f};
  v8f acc1 = {0.f, 0.f, 0.f, 0.f, 0.f, 0.f, 0.f, 0.f};
  {
    const int rb = row0 + ((lane >> 4) << 3);
    const int bc0 = (col0 + mloc < N)      ? (col0 + mloc)      : (N - 1);
    const int bc1 = (col0 + 16 + mloc < N) ? (col0 + 16 + mloc) : (N - 1);
    for (int k = 0; k < Kd; ++k) {
      const float bv0 = B[(long)k * rsB + bc0];
      const float bv1 = B[(long)k * rsB + bc1];
#pragma unroll
      for (int v = 0; v < 8; ++v) {
        const int rr = (rb + v < M) ? (rb + v) : (M - 1);
        const float av = A[(long)rr * rsA + k];
        acc0[v] += av * bv0;
        acc1[v] += av * bv1;
      }
    }
  }
#endif

  // --- store D (bounds-checked; ragged edges suppressed here) ---
  const int cn0 = col0 + mloc;
  const int cn1 = col0 + 16 + mloc;
  const int rbase = row0 + ((lane >> 4) << 3);
  if (cn0 < N) {
#pragma unroll
    for (int v = 0; v < 8; ++v) {
      const int rr = rbase + v;
      if (rr < M) {
        const long idx = (long)rr * rsC + (long)cn0 * csC;
        const float val = alpha * acc0[v];
        C[idx] = accum ? (C[idx] + val) : val;
      }
    }
  }
  if (cn1 < N) {
#pragma unroll
    for (int v = 0; v < 8; ++v) {
      const int rr = rbase + v;
      if (rr < M) {
        const long idx = (long)rr * rsC + (long)cn1 * csC;
        const float val = alpha * acc1[v];
        C[idx] = accum ? (C[idx] + val) : val;
      }
    }
  }
}

// ---------------------------------------------------------------------------
// Wavelet decompose: xa = [x[::2], x[1::2]] (last dim 2K), S = xa@ec_s,
// D = xa@ec_d. Reads reference layout (level 0) or channel-major (2048, n_in),
// writes channel-major (2048, n_out) so DFT-GEMM rows are contiguous.
// ---------------------------------------------------------------------------
__global__ __launch_bounds__(256) void mwno_decompose(
    const float* __restrict__ x, int ref_layout,
    const float* __restrict__ ec_s, const float* __restrict__ ec_d,
    float* __restrict__ S, float* __restrict__ D, int n_in)
{
  const int n_out = n_in >> 1;
  const long idx = (long)blockIdx.x * blockDim.x + threadIdx.x;
  const long total = (long)BQ * n_out * CQ;
  if (idx >= total) return;
  const int c = (int)(idx % CQ);
  const long r = idx / CQ;
  const int t = (int)(r % n_out);
  const int b = (int)(r / n_out);

  float xa[8];
  if (ref_layout) {
    const float* p0 = x + (((long)b * n_in + 2 * t    ) * CQ + c) * KQ;
    const float* p1 = x + (((long)b * n_in + 2 * t + 1) * CQ + c) * KQ;
#pragma unroll
    for (int kk = 0; kk < KQ; ++kk) { xa[kk] = p0[kk]; xa[KQ + kk] = p1[kk]; }
  } else {
    const long rowb = (long)b * CKQ + c * KQ;
#pragma unroll
    for (int kk = 0; kk < KQ; ++kk) {
      const float* rp = x + (rowb + kk) * n_in;
      xa[kk]      = rp[2 * t];
      xa[KQ + kk] = rp[2 * t + 1];
    }
  }
  const long ob = (long)b * CKQ + c * KQ;
#pragma unroll
  for (int ko = 0; ko < KQ; ++ko) {
    float sv = 0.f, dv = 0.f;
#pragma unroll
    for (int j = 0; j < 2 * KQ; ++j) {
      sv += xa[j] * ec_s[j * KQ + ko];
      dv += xa[j] * ec_d[j * KQ + ko];
    }
    S[(ob + ko) * n_out + t] = sv;
    D[(ob + ko) * n_out + t] = dv;
  }
}

// ---------------------------------------------------------------------------
// Reconstruct one level: xx = [x + Us, Ud]; even = xx@rc_e, odd = xx@rc_o,
// interleave. Final level writes straight into d_out in reference layout.
// ---------------------------------------------------------------------------
__global__ __launch_bounds__(256) void mwno_reconstruct(
    const float* __restrict__ x, const float* __restrict__ Us,
    const float* __restrict__ Ud,
    const float* __restrict__ rc_e, const float* __restrict__ rc_o,
    float* __restrict__ xout, int n, int to_ref)
{
  const long idx = (long)blockIdx.x * blockDim.x + threadIdx.x;
  const long total = (long)BQ * n * CQ;
  if (idx >= total) return;
  const int c = (int)(idx % CQ);
  const long r = idx / CQ;
  const int t = (int)(r % n);
  const int b = (int)(r / n);
  const long rowb = (long)b * CKQ + c * KQ;

  float xx[8];
#pragma unroll
  for (int kk = 0; kk < KQ; ++kk) {
    const long p = (rowb + kk) * n + t;
    xx[kk]      = x[p] + Us[p];
    xx[KQ + kk] = Ud[p];
  }
#pragma unroll
  for (int ko = 0; ko < KQ; ++ko) {
    float e = 0.f, o = 0.f;
#pragma unroll
    for (int j = 0; j < 2 * KQ; ++j) {
      e += xx[j] * rc_e[j * KQ + ko];
      o += xx[j] * rc_o[j * KQ + ko];
    }
    if (to_ref) {
      xout[(((long)b * (2 * n) + 2 * t    ) * CQ + c) * KQ + ko] = e;
      xout[(((long)b * (2 * n) + 2 * t + 1) * CQ + c) * KQ + ko] = o;
    } else {
      const long ob = (rowb + ko) * (2l * n);
      xout[ob + 2 * t]     = e;
      xout[ob + 2 * t + 1] = o;
    }
  }
}

// Forward partial-DFT twiddles: Fr[t*l+m]=cos(2*pi*t*m/n), Fi=-sin(...)
__global__ __launch_bounds__(256) void mwno_twiddle_fwd(
    float* __restrict__ Fr, float* __restrict__ Fi, int n, int l)
{
  const long idx = (long)blockIdx.x * blockDim.x + threadIdx.x;
  const long total = (long)n * l;
  if (idx >= total) return;
  const int m = (int)(idx % l);
  const int t = (int)(idx / l);
  const long tm = ((long)t * m) % n;
  const float ang = (float)(6.283185307179586476925286766559 * (double)tm / (double)n);
  Fr[idx] = cosf(ang);
  Fi[idx] = -sinf(ang);
}

// Inverse (irfft with hermitian fold): y_t = sum_m s_m*(XR*cos - XI*sin),
// s_m = 1/n for m==0 or m==n/2, else 2/n.  Gr pairs with XR, Gi with XI.
__global__ __launch_bounds__(256) void mwno_twiddle_inv(
    float* __restrict__ Gr, float* __restrict__ Gi, int n, int l)
{
  const long idx = (long)blockIdx.x * blockDim.x + threadIdx.x;
  const long total = (long)l * n;
  if (idx >= total) return;
  const int t = (int)(idx % n);
  const int m = (int)(idx / n);
  const long tm = ((long)t * m) % n;
  const float ang = (float)(6.283185307179586476925286766559 * (double)tm / (double)n);
  const float s = ((m == 0) || (2 * m == n)) ? (1.0f / (float)n) : (2.0f / (float)n);
  Gr[idx] = s * cosf(ang);
  Gi[idx] = -s * sinf(ang);
}

// Repack spectral weight (i,o,m) -> mode-major (m,i,o) for coalesced B operands.
__global__ __launch_bounds__(256) void mwno_repack(
    const float* __restrict__ W, float* __restrict__ Wp)
{
  const long idx = (long)blockIdx.x * blockDim.x + threadIdx.x;
  if (idx >= (long)CKQ * CKQ * ALPHAQ) return;
  const int o = (int)(idx & 127);
  const int i = (int)((idx >> 7) & 127);
  const int m = (int)(idx >> 14);
  Wp[idx] = W[((long)i * CKQ + o) * ALPHAQ + m];
}

// Coarsest-scale T0: x <- x @ T0_w^T + T0_b on the (2048, 1) buffer.
__global__ __launch_bounds__(256) void mwno_t0(
    float* __restrict__ x, const float* __restrict__ T0w,
    const float* __restrict__ T0b)
{
  const int idx = blockIdx.x * blockDim.x + threadIdx.x;
  if (idx >= BQ * CQ) return;
  const int c = idx & (CQ - 1);
  const int b = idx >> 5;
  const long base = (long)b * CKQ + c * KQ;
  float v[KQ];
#pragma unroll
  for (int kk = 0; kk < KQ; ++kk) v[kk] = x[base + kk];
#pragma unroll
  for (int ko = 0; ko < KQ; ++ko) {
    float s = T0b[ko];
#pragma unroll
    for (int kk = 0; kk < KQ; ++kk) s += v[kk] * T0w[ko * KQ + kk];
    x[base + ko] = s;
  }
}

// ---------------------------------------------------------------------------
extern "C" void kernel_launch(void* const* d_in, const int* in_sizes, int n_in,
                              void* d_out, int out_size, void* d_ws, size_t ws_size,
                              hipStream_t stream)
{
  (void)in_sizes; (void)n_in; (void)out_size;

  const float* x_in = (const float*)d_in[0];
  const float* wr[3] = {(const float*)d_in[1], (const float*)d_in[3], (const float*)d_in[5]};
  const float* wi[3] = {(const float*)d_in[2], (const float*)d_in[4], (const float*)d_in[6]};
  const float* ec_s = (const float*)d_in[7];
  const float* ec_d = (const float*)d_in[8];
  const float* rc_e = (const float*)d_in[9];
  const float* rc_o = (const float*)d_in[10];
  const float* T0w = (const float*)d_in[11];
  const float* T0b = (const float*)d_in[12];
  float* out = (float*)d_out;

  // ---- workspace carve-up (floats) ----
  float* ws = (float*)d_ws;
  size_t off = 0;
  auto take = [&](size_t nf) { float* p = ws + off; off += nf; return p; };
  const size_t WSZ = (size_t)CKQ * CKQ * ALPHAQ;      // 2M floats per weight
  float* wpr[3]; float* wpi[3];
  for (int i = 0; i < 3; ++i) { wpr[i] = take(WSZ); wpi[i] = take(WSZ); }
  float* S0 = take(2048l * 4096);
  float* S1 = take(2048l * 4096);
  float* Db = take(2048l * 4096);
  float* Fr = take(4096l * 128);
  float* Fi = take(4096l * 128);
  float* Gr = take(4096l * 128);
  float* Gi = take(4096l * 128);
  float* XRd = take(128l * 2048);
  float* XId = take(128l * 2048);
  float* XRs = take(128l * 2048);
  float* XIs = take(128l * 2048);
  float* UdfR = take(2048l * 128);
  float* UdfI = take(2048l * 128);
  float* UsfR = take(2048l * 128);
  float* UsfI = take(2048l * 128);
  float* UdT = take(2048l * 8191);
  float* UsT = take(2048l * 8191);
  if (off * sizeof(float) > ws_size) return;   // deterministic no-op if ws too small

  auto blocks1d = [](long total) { return (unsigned)((total + 255) / 256); };

  auto gemm = [&](const float* A, long rsA, long bsA,
                  const float* B, long rsB, long bsB,
                  float* C, long rsC, long csC, long bsC,
                  int M, int N, int Kd, float alpha, int accum, int batch)
  {
    const long tiles = (long)((M + 15) / 16) * ((N + 31) / 32);
    dim3 grid((unsigned)((tiles + 7) / 8), 1, (unsigned)batch);
    mwno_gemm<<<grid, dim3(256), 0, stream>>>(A, rsA, bsA, B, rsB, bsB,
                                              C, rsC, csC, bsC, M, N, Kd, alpha, accum);
  };

  // ---- one-time weight repack to mode-major (m, i, o) ----
  for (int i = 0; i < 3; ++i) {
    mwno_repack<<<blocks1d((long)WSZ), 256, 0, stream>>>(wr[i], wpr[i]);
    mwno_repack<<<blocks1d((long)WSZ), 256, 0, stream>>>(wi[i], wpi[i]);
  }

  // ---- level geometry ----
  int nlev[NLEV], llev[NLEV];
  long udoff[NLEV];
  long accu = 0;
  for (int j = 0; j < NLEV; ++j) {
    nlev[j] = NQ >> (j + 1);
    const int half = nlev[j] / 2 + 1;
    llev[j] = half < ALPHAQ ? half : ALPHAQ;
    udoff[j] = accu;
    accu += 2048l * nlev[j];
  }

  // ================= decomposition + spectral mixing =================
  for (int j = 0; j < NLEV; ++j) {
    const int n_inl = (j == 0) ? NQ : nlev[j - 1];
    const int n = nlev[j];
    const int l = llev[j];
    const float* xin = (j == 0) ? x_in : ((j & 1) ? S0 : S1);
    float* Sout = (j & 1) ? S1 : S0;

    mwno_decompose<<<blocks1d((long)BQ * n * CQ), 256, 0, stream>>>(
        xin, j == 0 ? 1 : 0, ec_s, ec_d, Sout, Db, n_inl);

    mwno_twiddle_fwd<<<blocks1d((long)n * l), 256, 0, stream>>>(Fr, Fi, n, l);
    mwno_twiddle_inv<<<blocks1d((long)l * n), 256, 0, stream>>>(Gr, Gi, n, l);

    // Forward partial DFT (rfft modes 0..l-1) as GEMM; outputs mode-major
    // X[m*2048 + b*128 + i]  (store strides rsC=1, csC=2048).
    gemm(Db,   n, 0, Fr, l, 0, XRd, 1, 2048, 0, 2048, l, n, 1.f, 0, 1);
    gemm(Db,   n, 0, Fi, l, 0, XId, 1, 2048, 0, 2048, l, n, 1.f, 0, 1);
    gemm(Sout, n, 0, Fr, l, 0, XRs, 1, 2048, 0, 2048, l, n, 1.f, 0, 1);
    gemm(Sout, n, 0, Fi, l, 0, XIs, 1, 2048, 0, 2048, l, n, 1.f, 0, 1);

    // Per-mode complex channel mixing, batched over modes (grid.z = l).
    // A: (16 x 128) @ X[m]   B: (128 x 128) @ Wp[m]   C row-major (2048, l).
    const long rA = 128, bA = 2048;
    const long rB = 128, bB = (long)CKQ * CKQ;
    const long rC = 128l * l, cC = l, bC = 1;
    // Ud_f = fft(d)*WA + fft(s)*WB  (complex)
    gemm(XRd, rA, bA, wpr[0], rB, bB, UdfR, rC, cC, bC, 16, 128, 128, +1.f, 0, l);
    gemm(XId, rA, bA, wpi[0], rB, bB, UdfR, rC, cC, bC, 16, 128, 128, -1.f, 1, l);
    gemm(XRs, rA, bA, wpr[1], rB, bB, UdfR, rC, cC, bC, 16, 128, 128, +1.f, 1, l);
    gemm(XIs, rA, bA, wpi[1], rB, bB, UdfR, rC, cC, bC, 16, 128, 128, -1.f, 1, l);

    gemm(XRd, rA, bA, wpi[0], rB, bB, UdfI, rC, cC, bC, 16, 128, 128, +1.f, 0, l);
    gemm(XId, rA, bA, wpr[0], rB, bB, UdfI, rC, cC, bC, 16, 128, 128, +1.f, 1, l);
    gemm(XRs, rA, bA, wpi[1], rB, bB, UdfI, rC, cC, bC, 16, 128, 128, +1.f, 1, l);
    gemm(XIs, rA, bA, wpr[1], rB, bB, UdfI, rC, cC, bC, 16, 128, 128, +1.f, 1, l);
    // Us_f = fft(d)*WC  (complex)
    gemm(XRd, rA, bA, wpr[2], rB, bB, UsfR, rC, cC, bC, 16, 128, 128, +1.f, 0, l);
    gemm(XId, rA, bA, wpi[2], rB, bB, UsfR, rC, cC, bC, 16, 128, 128, -1.f, 1, l);
    gemm(XRd, rA, bA, wpi[2], rB, bB, UsfI, rC, cC, bC, 16, 128, 128, +1.f, 0, l);
    gemm(XId, rA, bA, wpr[2], rB, bB, UsfI, rC, cC, bC, 16, 128, 128, +1.f, 1, l);

    // Inverse partial DFT back to time domain (channel-major (2048, n)).
    gemm(UdfR, l, 0, Gr, n, 0, UdT + udoff[j], n, 1, 0, 2048, n, l, 1.f, 0, 1);
    gemm(UdfI, l, 0, Gi, n, 0, UdT + udoff[j], n, 1, 0, 2048, n, l, 1.f, 1, 1);
    gemm(UsfR, l, 0, Gr, n, 0, UsT + udoff[j], n, 1, 0, 2048, n, l, 1.f, 0, 1);
    gemm(UsfI, l, 0, Gi, n, 0, UsT + udoff[j], n, 1, 0, 2048, n, l, 1.f, 1, 1);
  }

  // ================= coarsest-scale T0 (final S is in S0: 13 levels, last even)
  mwno_t0<<<blocks1d(BQ * CQ), 256, 0, stream>>>(S0, T0w, T0b);

  // ================= reconstruction =================
  for (int i = NLEV - 1; i >= 0; --i) {
    const int n = nlev[i];
    const int t = NLEV - 1 - i;                      // step index, t=0 first
    const float* xi = (t % 2 == 0) ? S0 : S1;        // ping-pong
    float* xo = (i == 0) ? out : ((t % 2 == 0) ? S1 : S0);
    mwno_reconstruct<<<blocks1d((long)BQ * n * CQ), 256, 0, stream>>>(
        xi, UsT + udoff[i], UdT + udoff[i], rc_e, rc_o, xo, n, (i == 0) ? 1 : 0);
  }
}